// BatchHardTripletLoss_2293512536516
// MI455X (gfx1250) — compile-verified
//
#include <hip/hip_runtime.h>
#include <hip/hip_bf16.h>

typedef __attribute__((ext_vector_type(2))) float v2f;
typedef __attribute__((ext_vector_type(4))) float v4f;
typedef __attribute__((ext_vector_type(8))) float v8f;

#define BATCH 8192
#define DDIM 128
#define NBLK (BATCH / 16)             // 512 row blocks
#define JSPLITS 8
#define JB_PER_SPLIT (NBLK / JSPLITS) // 64 j-blocks per split
#define MARGIN_F 1.0f

// ---------------------------------------------------------------------------
// Kernel 1: squared norms, one wave per row (lane loads float4 of 4 cols)
// ---------------------------------------------------------------------------
__global__ __launch_bounds__(256) void sqnorm_kernel(const float* __restrict__ E,
                                                     float* __restrict__ sq) {
    const int lane = threadIdx.x & 31;
    const int wid  = threadIdx.x >> 5;
    const int row  = blockIdx.x * 8 + wid;
    v4f v = *(const v4f*)(E + (size_t)row * DDIM + lane * 4);
    float s = v.x * v.x + v.y * v.y + v.z * v.z + v.w * v.w;
    s += __shfl_xor(s, 16, 32);
    s += __shfl_xor(s, 8, 32);
    s += __shfl_xor(s, 4, 32);
    s += __shfl_xor(s, 2, 32);
    s += __shfl_xor(s, 1, 32);
    if (lane == 0) sq[row] = s;
}

// ---------------------------------------------------------------------------
// Kernel 2: fused Gram (WMMA f32 16x16x4) + masked hardest-pos/neg reduction.
// Workgroup = 8 waves, each wave owns one 16-anchor i-block (A in 64 VGPRs),
// all waves share LDS-staged 16x128 B-tiles over one j-split. B tile is
// prefetched fully into registers per j-block; dual accumulators break the
// serial wmma dependency chain.
// ---------------------------------------------------------------------------
__global__ __launch_bounds__(256) void batchhard_kernel(
    const float* __restrict__ E, const int* __restrict__ labels,
    const float* __restrict__ sq,
    float* __restrict__ hp_part, float* __restrict__ hn_part) {
    __shared__ float tileB[16 * DDIM];  // 8 KB

    const int lane = threadIdx.x & 31;
    const int wid  = threadIdx.x >> 5;
    const int half = lane >> 4;   // K-half (ISA A/B layout)
    const int l16  = lane & 15;

    const int ib = blockIdx.x * 8 + wid;  // i-block 0..511
    const int i0 = ib * 16;
    const int split = blockIdx.y;

    // A-tile resident in registers: lane holds row (l16), K-pair 2*half+{0,1}
    v2f a[32];
    const float* arow = E + (size_t)(i0 + l16) * DDIM + 2 * half;
    #pragma unroll
    for (int s = 0; s < 32; ++s) a[s] = *(const v2f*)(arow + 4 * s);

    float sqi[8];
    int labi[8];
    #pragma unroll
    for (int v = 0; v < 8; ++v) {
        const int r = i0 + v + 8 * half;  // C/D layout: VGPR v <-> row v+8*half
        sqi[v]  = sq[r];
        labi[v] = labels[r];
    }

    float hp[8], hn[8];
    #pragma unroll
    for (int v = 0; v < 8; ++v) { hp[v] = -__builtin_inff(); hn[v] = __builtin_inff(); }

    const int jb0 = split * JB_PER_SPLIT;
    for (int jb = jb0; jb < jb0 + JB_PER_SPLIT; ++jb) {
        const int j0 = jb * 16;

        __syncthreads();
        {   // cooperative stage of B-tile: 256 threads x 8 floats (2x b128)
            const int t   = threadIdx.x;
            const int row = t >> 4;
            const int col = (t & 15) * 8;
            const float* src = E + (size_t)(j0 + row) * DDIM + col;
            float* dst = tileB + row * DDIM + col;
            *(v4f*)(dst)     = *(const v4f*)(src);
            *(v4f*)(dst + 4) = *(const v4f*)(src + 4);
        }
        __syncthreads();

        const float sqj = sq[j0 + l16];
        const int  labj = labels[j0 + l16];

        // prefetch the whole B tile for this wave into registers (64 VGPRs)
        v2f breg[32];
        const float* brow = tileB + l16 * DDIM + 2 * half;
        #pragma unroll
        for (int s = 0; s < 32; ++s) breg[s] = *(const v2f*)(brow + 4 * s);

        // dual-accumulator wmma chain: two independent dependency streams
        v8f acc0 = {}, acc1 = {};
        #pragma unroll
        for (int s = 0; s < 32; s += 2) {
            acc0 = __builtin_amdgcn_wmma_f32_16x16x4_f32(
                false, a[s],     false, breg[s],     (short)0, acc0, false, false);
            acc1 = __builtin_amdgcn_wmma_f32_16x16x4_f32(
                false, a[s + 1], false, breg[s + 1], (short)0, acc1, false, false);
        }

        #pragma unroll
        for (int v = 0; v < 8; ++v) {
            const float g  = acc0[v] + acc1[v];
            const float d2 = sqi[v] + sqj - 2.0f * g;
            const float d  = sqrtf(fmaxf(d2, 0.0f));
            if (labi[v] == labj) hp[v] = fmaxf(hp[v], d);
            else                 hn[v] = fminf(hn[v], d);
        }
    }

    // reduce hardest-pos/neg across the 16 columns held in each lane half
    #pragma unroll
    for (int v = 0; v < 8; ++v) {
        #pragma unroll
        for (int m = 8; m >= 1; m >>= 1) {
            hp[v] = fmaxf(hp[v], __shfl_xor(hp[v], m, 16));
            hn[v] = fminf(hn[v], __shfl_xor(hn[v], m, 16));
        }
    }
    if (l16 == 0) {
        #pragma unroll
        for (int v = 0; v < 8; ++v) {
            const int r = i0 + v + 8 * half;
            hp_part[(size_t)split * BATCH + r] = hp[v];
            hn_part[(size_t)split * BATCH + r] = hn[v];
        }
    }
}

// ---------------------------------------------------------------------------
// Kernel 3: combine j-splits, relu(hp - hn + margin), mean over batch
// ---------------------------------------------------------------------------
__global__ __launch_bounds__(256) void finalize_kernel(
    const float* __restrict__ hp_part, const float* __restrict__ hn_part,
    float* __restrict__ out) {
    __shared__ float red[256];
    float acc = 0.0f;
    for (int r = threadIdx.x; r < BATCH; r += 256) {
        float hp = -__builtin_inff(), hn = __builtin_inff();
        #pragma unroll
        for (int s = 0; s < JSPLITS; ++s) {
            hp = fmaxf(hp, hp_part[s * BATCH + r]);
            hn = fminf(hn, hn_part[s * BATCH + r]);
        }
        acc += fmaxf(hp - hn + MARGIN_F, 0.0f);
    }
    red[threadIdx.x] = acc;
    __syncthreads();
    for (int s = 128; s > 0; s >>= 1) {
        if (threadIdx.x < s) red[threadIdx.x] += red[threadIdx.x + s];
        __syncthreads();
    }
    if (threadIdx.x == 0) out[0] = red[0] / (float)BATCH;
}

// ---------------------------------------------------------------------------
extern "C" void kernel_launch(void* const* d_in, const int* in_sizes, int n_in,
                              void* d_out, int out_size, void* d_ws, size_t ws_size,
                              hipStream_t stream) {
    const float* E      = (const float*)d_in[0];
    const int*   labels = (const int*)d_in[1];

    float* sq      = (float*)d_ws;                 // [8192]
    float* hp_part = sq + BATCH;                   // [8][8192]
    float* hn_part = hp_part + JSPLITS * BATCH;    // [8][8192]
    float* out     = (float*)d_out;

    sqnorm_kernel<<<BATCH / 8, 256, 0, stream>>>(E, sq);

    dim3 grid(NBLK / 8, JSPLITS);  // 64 i-groups x 8 j-splits = 512 WGs
    batchhard_kernel<<<grid, 256, 0, stream>>>(E, labels, sq, hp_part, hn_part);

    finalize_kernel<<<1, 256, 0, stream>>>(hp_part, hn_part, out);
}